// BilinearDistance_10385230922578
// MI455X (gfx1250) — compile-verified
//
#include <hip/hip_runtime.h>

// ---------------------------------------------------------------------------
// BilinearDistance on MI455X (gfx1250): fused two-GEMM, fp32 WMMA, wave32.
//   dists[b,n,m] = sum_{i,j} data[b,n,i] * W[i,j] * crit[b,m,j]
// Store-bound (268 MB out vs 18.3 GFLOP): keep fp32, fuse both GEMMs in LDS.
// ---------------------------------------------------------------------------

typedef __attribute__((ext_vector_type(2))) float v2f;
typedef __attribute__((ext_vector_type(8))) float v8f;

#define BB 16
#define NN 2048
#define MM 2048
#define DD 128

#define TILE_N 128   // output rows (data side) per workgroup
#define TILE_M 256   // output cols (crit side) per workgroup
#define LDA 132      // padded pitch (dwords): 132 % 64 == 4 -> conflict-free b64 frag reads
#define LDB 132
#define LDW 136      // 136 % 64 == 8 -> conflict-free strided b32 reads of W columns

__global__ __launch_bounds__(256)
void bilinear_wmma_fused(const float* __restrict__ data,
                         const float* __restrict__ crit,
                         const float* __restrict__ W,
                         float* __restrict__ out) {
  extern __shared__ float smem[];
  float* sA = smem;                  // 128 x 132 : data tile, then lw tile (in-place)
  float* sB = sA + TILE_N * LDA;     // 256 x 132 : crit tile
  float* sW = sB + TILE_M * LDB;     // 128 x 136 : full W

  const int tid  = threadIdx.x;
  const int lane = tid & 31;
  const int wv   = tid >> 5;       // wave 0..7
  const int w_m  = wv & 3;         // 4 waves along M
  const int w_n  = wv >> 2;        // 2 waves along N
  const int lo   = lane & 15;      // column / row-within-subtile lane index
  const int hi   = lane >> 4;      // K-half selector

  const int mblk = blockIdx.x;
  const int nblk = blockIdx.y;
  const int b    = blockIdx.z;

  // ---------------- cooperative staging: global -> LDS (float4) -------------
  {
    const float* gA = data + ((size_t)b * NN + (size_t)nblk * TILE_N) * DD;
    for (int idx = tid; idx < TILE_N * (DD / 4); idx += 256) {
      int r = idx >> 5, c = idx & 31;
      *(float4*)&sA[r * LDA + c * 4] = *(const float4*)&gA[r * DD + c * 4];
    }
    const float* gB = crit + ((size_t)b * MM + (size_t)mblk * TILE_M) * DD;
    for (int idx = tid; idx < TILE_M * (DD / 4); idx += 256) {
      int r = idx >> 5, c = idx & 31;
      *(float4*)&sB[r * LDB + c * 4] = *(const float4*)&gB[r * DD + c * 4];
    }
    for (int idx = tid; idx < DD * (DD / 4); idx += 256) {
      int r = idx >> 5, c = idx & 31;
      *(float4*)&sW[r * LDW + c * 4] = *(const float4*)&W[r * DD + c * 4];
    }
  }
  __syncthreads();

  // ---------------- stage 1: lw_tile(128x128) = data_tile @ W ---------------
  // Wave owns 64(N) x 32(J): 4x2 subtiles of 16x16, K = 128 in steps of 4.
  {
    v8f acc1[4][2];
#pragma unroll
    for (int i = 0; i < 4; ++i)
#pragma unroll
      for (int j = 0; j < 2; ++j)
        acc1[i][j] = (v8f){0.f, 0.f, 0.f, 0.f, 0.f, 0.f, 0.f, 0.f};

    const int aRow = 64 * w_n + lo;   // A: lane L%16 = row
    const int jCol = 32 * w_m + lo;   // B: lane L%16 = column (j of W)
    for (int kk = 0; kk < DD; kk += 4) {
      const int k0 = kk + 2 * hi;     // A/B VGPR v holds k = 2*(L/16) + v
      v2f af[4], wf[2];
#pragma unroll
      for (int i = 0; i < 4; ++i)
        af[i] = *(const v2f*)&sA[(aRow + 16 * i) * LDA + k0];
#pragma unroll
      for (int j = 0; j < 2; ++j)
        wf[j] = (v2f){ sW[k0 * LDW + jCol + 16 * j],
                       sW[(k0 + 1) * LDW + jCol + 16 * j] };
#pragma unroll
      for (int i = 0; i < 4; ++i)
#pragma unroll
        for (int j = 0; j < 2; ++j)
          acc1[i][j] = __builtin_amdgcn_wmma_f32_16x16x4_f32(
              false, af[i], false, wf[j], (short)0, acc1[i][j], false, false);
    }

    __syncthreads();  // everyone done READING the data tile in sA
    // Write lw tile back into sA in row-major (row = n, col = j), padded pitch.
    // C/D layout: VGPR v -> row 16*i + v + 8*hi ; lane%16 -> column.
#pragma unroll
    for (int i = 0; i < 4; ++i)
#pragma unroll
      for (int j = 0; j < 2; ++j)
#pragma unroll
        for (int v = 0; v < 8; ++v)
          sA[(64 * w_n + 16 * i + v + 8 * hi) * LDA + 32 * w_m + 16 * j + lo]
              = acc1[i][j][v];
    __syncthreads();
  }

  // ---------------- stage 2: out_tile = lw_tile @ crit_tile^T ---------------
  // Wave owns 64(N) x 64(M): 4x4 subtiles, K = 128 in steps of 4.
  v8f acc[4][4];
#pragma unroll
  for (int i = 0; i < 4; ++i)
#pragma unroll
    for (int j = 0; j < 4; ++j)
      acc[i][j] = (v8f){0.f, 0.f, 0.f, 0.f, 0.f, 0.f, 0.f, 0.f};

  const int aRow2 = 64 * w_n + lo;
  const int bRow2 = 64 * w_m + lo;   // crit row m maps to the WMMA "column"
  for (int kk = 0; kk < DD; kk += 4) {
    const int k0 = kk + 2 * hi;
    v2f af[4], bf[4];
#pragma unroll
    for (int i = 0; i < 4; ++i)
      af[i] = *(const v2f*)&sA[(aRow2 + 16 * i) * LDA + k0];
#pragma unroll
    for (int j = 0; j < 4; ++j)
      bf[j] = *(const v2f*)&sB[(bRow2 + 16 * j) * LDB + k0];
#pragma unroll
    for (int i = 0; i < 4; ++i)
#pragma unroll
      for (int j = 0; j < 4; ++j)
        acc[i][j] = __builtin_amdgcn_wmma_f32_16x16x4_f32(
            false, af[i], false, bf[j], (short)0, acc[i][j], false, false);
  }

  // ---------------- store 64x64 per wave to global -------------------------
  float* outB = out + (size_t)b * NN * MM;
#pragma unroll
  for (int i = 0; i < 4; ++i) {
#pragma unroll
    for (int v = 0; v < 8; ++v) {
      const int n = nblk * TILE_N + 64 * w_n + 16 * i + v + 8 * hi;
      float* rp = outB + (size_t)n * MM + mblk * TILE_M + 64 * w_m;
#pragma unroll
      for (int j = 0; j < 4; ++j)
        rp[16 * j + lo] = acc[i][j][v];
    }
  }
}

extern "C" void kernel_launch(void* const* d_in, const int* in_sizes, int n_in,
                              void* d_out, int out_size, void* d_ws, size_t ws_size,
                              hipStream_t stream) {
  (void)in_sizes; (void)n_in; (void)out_size; (void)d_ws; (void)ws_size;
  const float* data = (const float*)d_in[0];  // [16, 2048, 128]
  const float* crit = (const float*)d_in[1];  // [16, 2048, 128]
  const float* W    = (const float*)d_in[2];  // [1, 128, 128]
  float* out = (float*)d_out;                 // [16, 2048, 2048]

  const size_t lds_bytes =
      (size_t)(TILE_N * LDA + TILE_M * LDB + DD * LDW) * sizeof(float);

  // Opt in to large dynamic LDS (gfx1250 allows up to 320 KB per workgroup).
  (void)hipFuncSetAttribute((const void*)bilinear_wmma_fused,
                            hipFuncAttributeMaxDynamicSharedMemorySize,
                            (int)lds_bytes);

  dim3 grid(MM / TILE_M, NN / TILE_N, BB);  // (8, 16, 16)
  bilinear_wmma_fused<<<grid, 256, lds_bytes, stream>>>(data, crit, W, out);
}